// BPICDetector__58583353917775
// MI455X (gfx1250) — compile-verified
//
#include <hip/hip_runtime.h>
#include <hip/hip_bf16.h>

typedef float v2f __attribute__((ext_vector_type(2)));
typedef float v8f __attribute__((ext_vector_type(8)));

#define BPIC_ITER 5
#define BPIC_MINVAR 1e-7f

__device__ __forceinline__ float softplusf(float x) {
    return (x > 20.f) ? x : log1pf(expf(x));
}
__device__ __forceinline__ float fast_rcp(float x) {
    return __builtin_amdgcn_rcpf(x);
}

// One wave32 per TWO batch items: lanes 0-15 -> item 2b, lanes 16-31 -> item 2b+1.
__global__ __launch_bounds__(32)
void bpic_kernel(const float* __restrict__ y_re, const float* __restrict__ y_im,
                 const float* __restrict__ h_re, const float* __restrict__ h_im,
                 const float* __restrict__ prior, const float* __restrict__ No_p,
                 float* __restrict__ out, int B)
{
    const int l = threadIdx.x;     // 0..31
    const int g = l >> 4;          // which item half (0/1)
    const int k = l & 15;          // symbol / row index owned by this lane
    int ib = blockIdx.x * 2 + g;   // item for the scalar phase
    if (ib > B - 1) ib = B - 1;    // uniform data clamp, EXEC stays full

    const float No = No_p[0];
    const float invNo = fast_rcp(No);

    __shared__ float2 sH[2][256];  // HtH (scaled by 1/No), (re,im)
    __shared__ float2 sG[2][256];  // G = HtH + No*I -> Z = G^-1 after GJ
    __shared__ float2 sX[2][16];   // vector staging

    // ---------------- HtH via V_WMMA_F32_16X16X4_F32, two items ----------------
    // A = H^T slice (16x4), B = H slice (4x16): identical lane register contents.
    const int off = g * 2;         // K-slot pair for this half-wave
#pragma unroll
    for (int q = 0; q < 2; ++q) {
        int item = blockIdx.x * 2 + q;
        if (item > B - 1) item = B - 1;
        const float* hrq = h_re + (size_t)item * 512;   // [m][k], M=32, K=16
        const float* hiq = h_im + (size_t)item * 512;
        v8f acc_re = {0.f,0.f,0.f,0.f,0.f,0.f,0.f,0.f};
        v8f acc_im = acc_re;
#pragma unroll
        for (int c = 0; c < 8; ++c) {
            const int m0 = c * 4;
            v2f ar, ai, nai;
            ar.x = hrq[(m0 + off    ) * 16 + k];
            ar.y = hrq[(m0 + off + 1) * 16 + k];
            ai.x = hiq[(m0 + off    ) * 16 + k];
            ai.y = hiq[(m0 + off + 1) * 16 + k];
            nai.x = -ai.x; nai.y = -ai.y;
            // Re += Hr^T Hr + Hi^T Hi
            acc_re = __builtin_amdgcn_wmma_f32_16x16x4_f32(false, ar, false, ar, (short)0, acc_re, false, false);
            acc_re = __builtin_amdgcn_wmma_f32_16x16x4_f32(false, ai, false, ai, (short)0, acc_re, false, false);
            // Im += Hr^T Hi - Hi^T Hr
            acc_im = __builtin_amdgcn_wmma_f32_16x16x4_f32(false, ar,  false, ai, (short)0, acc_im, false, false);
            acc_im = __builtin_amdgcn_wmma_f32_16x16x4_f32(false, nai, false, ar, (short)0, acc_im, false, false);
        }
        // C/D layout: vgpr r, lanes 0-15 -> (M=r, N=lane); lanes 16-31 -> (M=r+8, N=lane-16)
        const int mbase = g * 8;
#pragma unroll
        for (int r = 0; r < 8; ++r) {
            int m = mbase + r;
            float re = acc_re[r] * invNo;      // HtH of whitened H = raw / No
            float im = acc_im[r] * invNo;
            sH[q][m * 16 + k] = make_float2(re, im);
            sG[q][m * 16 + k] = make_float2(re + ((m == k) ? No : 0.f), im); // reg = No/es, es=1
        }
    }
    __syncthreads();

    // ---------------- Hty (lane (g,k) owns row k of item ib) ----------------
    const float* hr = h_re + (size_t)ib * 512;
    const float* hi = h_im + (size_t)ib * 512;
    float tyr = 0.f, tyi = 0.f;
    {
        const float* yrp = y_re + (size_t)ib * 32;
        const float* yip = y_im + (size_t)ib * 32;
#pragma unroll 4
        for (int m = 0; m < 32; ++m) {
            float yr = yrp[m], yi = yip[m];
            float a = hr[m * 16 + k], c = hi[m * 16 + k];
            tyr += a * yr + c * yi;            // conj(h) * y
            tyi += a * yi - c * yr;
        }
        tyr *= invNo; tyi *= invNo;
    }

    // ------- In-place complex Gauss-Jordan: each half-wave inverts its own G -------
    for (int kp = 0; kp < 16; ++kp) {
        float2 piv = sG[g][kp * 17];
        float dmag = fast_rcp(piv.x * piv.x + piv.y * piv.y);
        float ipr = piv.x * dmag, ipi = -piv.y * dmag;  // 1/pivot
        __syncthreads();
        {   // scale row kp unconditionally (lane k==kp writes ~1+0i; fixed in phase 3)
            float2 a = sG[g][kp * 16 + k];
            sG[g][kp * 16 + k] = make_float2(a.x * ipr - a.y * ipi,
                                             a.x * ipi + a.y * ipr);
        }
        __syncthreads();
        float2 f = sG[g][k * 16 + kp];                  // old column entry (factor)
        if (k != kp) {                                  // eliminate row k, all 16 cols
#pragma unroll
            for (int j = 0; j < 16; ++j) {
                float2 akj = sG[g][kp * 16 + j];
                float2 v = sG[g][k * 16 + j];
                v.x -= f.x * akj.x - f.y * akj.y;
                v.y -= f.x * akj.y + f.y * akj.x;
                sG[g][k * 16 + j] = v;                  // j==kp garbage, overwritten below
            }
        }
        __syncthreads();
        if (k == kp) sG[g][kp * 17] = make_float2(ipr, ipi);
        else         sG[g][k * 16 + kp] = make_float2(-(f.x * ipr - f.y * ipi),
                                                      -(f.x * ipi + f.y * ipr));
        __syncthreads();
    }

    // ---------------- Constellation + prior logits ----------------
    // softplus factorization: only 8 distinct values softplus(+/-prior_t)
    float pre_[16], pim_[16], lp[16];
    {
        const float rs10 = 0.31622776601683794f;   // 1/sqrt(10): 16-QAM norm
        const float* pp = prior + ((size_t)ib * 16 + k) * 4;
        float pv[4] = { pp[0], pp[1], pp[2], pp[3] };
        float s1[4], s0[4];                        // bit=1: softplus(-p), bit=0: softplus(+p)
#pragma unroll
        for (int t = 0; t < 4; ++t) { s1[t] = softplusf(-pv[t]); s0[t] = softplusf(pv[t]); }
#pragma unroll
        for (int c = 0; c < 16; ++c) {
            int b0 = (c >> 3) & 1, b1 = (c >> 2) & 1, b2 = (c >> 1) & 1, b3 = c & 1;
            pre_[c] = (float)((1 - 2 * b0) * (2 - (1 - 2 * b2))) * rs10;
            pim_[c] = (float)((1 - 2 * b1) * (2 - (1 - 2 * b3))) * rs10;
            lp[c] = -((b0 ? s1[0] : s0[0]) + (b1 ? s1[1] : s0[1]) +
                      (b2 ? s1[2] : s0[2]) + (b3 ? s1[3] : s0[3]));
        }
    }

    // p0 = softmax(lp); x_dsc, v_dsc
    float xdr, xdi, vd;
    {
        float mx = lp[0];
#pragma unroll
        for (int c = 1; c < 16; ++c) mx = fmaxf(mx, lp[c]);
        float p0c[16], sum = 0.f;
#pragma unroll
        for (int c = 0; c < 16; ++c) { p0c[c] = __expf(lp[c] - mx); sum += p0c[c]; }
        float isum = fast_rcp(sum);
        float e2 = 0.f; xdr = 0.f; xdi = 0.f;
#pragma unroll
        for (int c = 0; c < 16; ++c) {
            float p = p0c[c] * isum;
            xdr += p * pre_[c]; xdi += p * pim_[c];
            e2  += p * (pre_[c] * pre_[c] + pim_[c] * pim_[c]);
        }
        vd = e2 - (xdr * xdr + xdi * xdi);
    }

    const float2 akk = sH[g][k * 17];               // diag(HtH) as stored
    const float invd = fast_rcp(akk.x);             // real > 0
    const float vbso = fmaxf(No * invd, BPIC_MINVAR);

    // ---------------- BPIC iterations ----------------
    float xpr = xdr, xpi = xdi, vp = vd, isep = 0.f;
#pragma unroll 1
    for (int it = 0; it < BPIC_ITER; ++it) {
        __syncthreads();
        sX[g][k] = make_float2(xdr, xdi);
        __syncthreads();
        // t = Hty - HtH @ x_dsc + akk*x_k   (== Hty - HtH_off @ x_dsc, exact)
        float tr = tyr + (akk.x * xdr - akk.y * xdi);
        float ti = tyi + (akk.x * xdi + akk.y * xdr);
#pragma unroll
        for (int j = 0; j < 16; ++j) {
            float2 a = sH[g][k * 16 + j];
            float2 x = sX[g][j];
            tr -= a.x * x.x - a.y * x.y;
            ti -= a.x * x.y + a.y * x.x;
        }
        float xbr, xbi;
        if (it == 0) {            // x_bso = Z @ t
            __syncthreads();
            sX[g][k] = make_float2(tr, ti);
            __syncthreads();
            xbr = 0.f; xbi = 0.f;
#pragma unroll
            for (int j = 0; j < 16; ++j) {
                float2 a = sG[g][k * 16 + j];
                float2 x = sX[g][j];
                xbr += a.x * x.x - a.y * x.y;
                xbi += a.x * x.y + a.y * x.x;
            }
        } else {                  // mrc: x_bso = t / diag
            xbr = tr * invd; xbi = ti * invd;
        }
        // BSE: softmax over 16 constellation points
        float sc[16], smx = -1e30f;
        const float nhalf = -0.5f * fast_rcp(vbso);
#pragma unroll
        for (int c = 0; c < 16; ++c) {
            float dr = xbr - pre_[c], di = xbi - pim_[c];
            sc[c] = (dr * dr + di * di) * nhalf;
            smx = fmaxf(smx, sc[c]);
        }
        float ssum = 0.f;
#pragma unroll
        for (int c = 0; c < 16; ++c) { sc[c] = __expf(sc[c] - smx); ssum += sc[c]; }
        float isum = fast_rcp(ssum);
        float xer = 0.f, xei = 0.f;
#pragma unroll
        for (int c = 0; c < 16; ++c) {
            float p = sc[c] * isum; sc[c] = p;
            xer += p * pre_[c]; xei += p * pim_[c];
        }
        float vbe = 0.f;
#pragma unroll
        for (int c = 0; c < 16; ++c) {
            float dr = pre_[c] - xer, di = pim_[c] - xei;
            vbe += sc[c] * (dr * dr + di * di);
        }
        vbe = fmaxf(vbe, BPIC_MINVAR);
        // ise = |mrc @ (Hty - HtH @ x_bse)|^2   (full matvec, incl. diagonal)
        __syncthreads();
        sX[g][k] = make_float2(xer, xei);
        __syncthreads();
        float rr = tyr, ri = tyi;
#pragma unroll
        for (int j = 0; j < 16; ++j) {
            float2 a = sH[g][k * 16 + j];
            float2 x = sX[g][j];
            rr -= a.x * x.x - a.y * x.y;
            ri -= a.x * x.y + a.y * x.x;
        }
        float ise = (rr * rr + ri * ri) * invd * invd;
        // DSC blend
        if (it == 0) { xdr = xer; xdi = xei; vd = vbe; }
        else {
            float rho = isep * fast_rcp(isep + ise + 1e-30f);
            xdr = (1.f - rho) * xpr + rho * xer;
            xdi = (1.f - rho) * xpi + rho * xei;
            vd  = (1.f - rho) * vp  + rho * vbe;
        }
        isep = ise; xpr = xer; xpi = xei; vp = vbe;
    }

    // ---------------- LLR output ----------------
    float ivout = fast_rcp(fmaxf(vd, BPIC_MINVAR));
    float best1[4] = {-1e30f, -1e30f, -1e30f, -1e30f};
    float best0[4] = {-1e30f, -1e30f, -1e30f, -1e30f};
#pragma unroll
    for (int c = 0; c < 16; ++c) {
        float dr = xdr - pre_[c], di = xdi - pim_[c];
        float q = -(dr * dr + di * di) * ivout + lp[c];
#pragma unroll
        for (int t = 0; t < 4; ++t) {
            if ((c >> (3 - t)) & 1) best1[t] = fmaxf(best1[t], q);
            else                    best0[t] = fmaxf(best0[t], q);
        }
    }
    {
        float* o = out + ((size_t)ib * 16 + k) * 4;
#pragma unroll
        for (int t = 0; t < 4; ++t) o[t] = best1[t] - best0[t];
    }
}

extern "C" void kernel_launch(void* const* d_in, const int* in_sizes, int n_in,
                              void* d_out, int out_size, void* d_ws, size_t ws_size,
                              hipStream_t stream) {
    const float* y_re  = (const float*)d_in[0];
    const float* y_im  = (const float*)d_in[1];
    const float* h_re  = (const float*)d_in[2];
    const float* h_im  = (const float*)d_in[3];
    const float* prior = (const float*)d_in[4];
    const float* No    = (const float*)d_in[5];
    const int B = in_sizes[0] / 32;        // y is (B, 32)
    const int nblk = (B + 1) / 2;          // two items per wave32
    bpic_kernel<<<nblk, 32, 0, stream>>>(y_re, y_im, h_re, h_im, prior, No,
                                         (float*)d_out, B);
}